// SingleHeadAttention_48945447306028
// MI455X (gfx1250) — compile-verified
//
#include <hip/hip_runtime.h>
#include <cstddef>
#include <cstdint>

// Single-head attention, MI455X (gfx1250), wave32.
// fp32 WMMA end-to-end (the TF-style multiplicative mask multiplies scores by
// -1e9; softmax then selects ~argmin of raw scores, so reduced precision would
// change the selected rows entirely).  All tile movement via the Tensor Data
// Mover (tensor_load_to_lds, TENSORcnt double-buffering).  W is pre-transposed
// and V is produced transposed so every WMMA operand pair is one aligned
// ds_load_b64 (no register-pairing movs).

typedef float v2f __attribute__((ext_vector_type(2)));
typedef float v8f __attribute__((ext_vector_type(8)));
typedef unsigned v4u __attribute__((ext_vector_type(4)));
typedef int v8i __attribute__((ext_vector_type(8)));
typedef int v4i __attribute__((ext_vector_type(4)));

#define SEQ   2048
#define EMB   1024
#define KDIM  128
#define BATCH 4

static __device__ __forceinline__ v8f wmma4(v2f a, v2f b, v8f c) {
  // V_WMMA_F32_16X16X4_F32 : D(16x16 f32) = A(16x4) x B(4x16) + C
  return __builtin_amdgcn_wmma_f32_16x16x4_f32(
      false, a, false, b, (short)0, c, false, false);
}

// Generic LDS pointer: addr[31:0] is the LDS byte offset (ISA 10.2 aperture map).
static __device__ __forceinline__ unsigned lds_addr32(const void* p) {
  return (unsigned)(uintptr_t)p;
}

// TDM: async copy a [rows x cols] fp32 tile (global row stride = stride_elems)
// into LDS, padding +4 dwords after every row (LDS row stride = cols + 4).
// pad_code: 4 -> every 32 dwords, 5 -> every 64, 6 -> every 128.
static __device__ __forceinline__ void tdm_load_tile_f32(
    const float* g, unsigned lds, unsigned rows, unsigned cols,
    unsigned stride_elems, unsigned pad_code)
{
  const unsigned long long ga = (unsigned long long)(uintptr_t)g;
  v4u g0;
  g0[0] = 1u;                                    // D# group0: count=1, user mode
  g0[1] = lds;                                   // lds_addr (bytes)
  g0[2] = (unsigned)ga;                          // global_addr[31:0]
  g0[3] = (unsigned)((ga >> 32) & 0x1FFFFFFull)  // global_addr[56:32]
        | (2u << 30);                            // type = 2 ("image")
  v8i g1;
  g1[0] = (int)((2u << 16)                       // data_size = 4 bytes
              | (1u << 20)                       // pad_enable (load only)
              | (pad_code << 22)                 // pad_interval
              | (3u << 25));                     // pad_amount = 4 dwords
  g1[1] = (int)((cols & 0xFFFFu) << 16);         // tensor_dim0[15:0]  (bit 48)
  g1[2] = (int)((cols >> 16) | ((rows & 0xFFFFu) << 16)); // td0 hi | td1 lo
  g1[3] = (int)((rows >> 16) | (cols << 16));    // td1 hi | tile_dim0
  g1[4] = (int)rows;                             // tile_dim1 (tile_dim2 = 0)
  g1[5] = (int)stride_elems;                     // tensor_dim0_stride[31:0]
  g1[6] = 0;
  g1[7] = 0;
  v4i z = {};
#if defined(__clang_major__) && (__clang_major__ >= 23)
  v8i z8 = {};
  __builtin_amdgcn_tensor_load_to_lds(g0, g1, z, z, z8, 0);
#else
  __builtin_amdgcn_tensor_load_to_lds(g0, g1, z, z, 0);
#endif
}

static __device__ __forceinline__ void wait_tensor_le2() {
  asm volatile("s_wait_tensorcnt 0x2" ::: "memory");
}
static __device__ __forceinline__ void wait_tensor_0() {
  asm volatile("s_wait_tensorcnt 0x0" ::: "memory");
}

// ---------------------------------------------------------------------------
// Kernel 0: transpose the three 1024x128 weight matrices -> Wt[128][1024].
// Output-coalesced; one-shot 1.5 MB.
// ---------------------------------------------------------------------------
__global__ __launch_bounds__(256) void transpose_w_kernel(
    const float* __restrict__ WQ, const float* __restrict__ WK,
    const float* __restrict__ WV,
    float* __restrict__ WQt, float* __restrict__ WKt, float* __restrict__ WVt)
{
  const int sel = blockIdx.y;
  const float* W  = (sel == 0) ? WQ  : (sel == 1) ? WK  : WV;
  float*       Wt = (sel == 0) ? WQt : (sel == 1) ? WKt : WVt;
  const int idx = blockIdx.x * 256 + threadIdx.x;   // over KDIM*EMB
  const int d = idx & (EMB - 1);
  const int c = idx >> 10;
  Wt[idx] = W[(size_t)d * KDIM + c];
}

// ---------------------------------------------------------------------------
// Kernel 1: projections.  One wave computes a 16x128 fp32 tile (K=1024).
// Input tile 16x64 (LDS stride 68) + Wt tile 128x64 (LDS stride 68), both via
// double-buffered TDM.  All WMMA operands are single aligned b64 LDS loads.
// sel==2 (V) stores its result transposed (vT[B][KD][S]) via an LDS staging
// transpose so the attention kernel gets consecutive PV operand pairs too.
// ---------------------------------------------------------------------------
__global__ __launch_bounds__(32) void proj_kernel(
    const float* __restrict__ qin, const float* __restrict__ kin,
    const float* __restrict__ vin,
    const float* __restrict__ WQt, const float* __restrict__ WKt,
    const float* __restrict__ WVt,
    float* __restrict__ qo, float* __restrict__ ko, float* __restrict__ vto)
{
  const int lane = threadIdx.x;
  const int ln   = lane & 15;
  const int g    = lane >> 4;
  const int sel  = blockIdx.y;

  const float* in = (sel == 0) ? qin : (sel == 1) ? kin : vin;
  const float* Wt = (sel == 0) ? WQt : (sel == 1) ? WKt : WVt;

  const size_t rowbase = (size_t)blockIdx.x * 16;

  __shared__ float atile[2][16][68];     // 16x64 input chunk, padded rows
  __shared__ float wtile[2][128][68];    // Wt 128x64 chunk, padded rows

  v8f acc[8] = {};

  // prologue: chunk 0 in flight
  tdm_load_tile_f32(in + rowbase * EMB, lds_addr32(&atile[0][0][0]),
                    16, 64, EMB, 5);
  tdm_load_tile_f32(Wt, lds_addr32(&wtile[0][0][0]), 128, 64, EMB, 5);

#pragma unroll 1
  for (int dc = 0; dc < EMB; dc += 64) {
    const int cur = (dc >> 6) & 1;
    if (dc + 64 < EMB) {
      tdm_load_tile_f32(in + rowbase * EMB + dc + 64,
                        lds_addr32(&atile[cur ^ 1][0][0]), 16, 64, EMB, 5);
      tdm_load_tile_f32(Wt + dc + 64,
                        lds_addr32(&wtile[cur ^ 1][0][0]), 128, 64, EMB, 5);
      wait_tensor_le2();   // chunk `cur` resident (in-order TDM completion)
    } else {
      wait_tensor_0();
    }

#pragma unroll
    for (int kk = 0; kk < 64; kk += 4) {
      const v2f a = *(const v2f*)&atile[cur][ln][kk + 2 * g];
#pragma unroll
      for (int t = 0; t < 8; ++t) {
        const v2f b = *(const v2f*)&wtile[cur][t * 16 + ln][kk + 2 * g];
        acc[t] = wmma4(a, b, acc[t]);
      }
    }
  }

  if (sel != 2) {
    // row-major store: lane holds (M = r + 8*g, N = ln)
    float* out = (sel == 0) ? qo : ko;
#pragma unroll
    for (int t = 0; t < 8; ++t) {
#pragma unroll
      for (int r = 0; r < 8; ++r) {
        const size_t row = rowbase + r + 8 * g;
        out[row * KDIM + t * 16 + ln] = acc[t][r];
      }
    }
  } else {
    // transposed store via LDS staging: vT[b][col][s]
    const int bi    = (int)(rowbase / SEQ);
    const int sbase = (int)(rowbase % SEQ);
    float* stg = &wtile[0][0][0];     // reuse as stage[16][132]
    __syncthreads();
#pragma unroll
    for (int t = 0; t < 8; ++t)
#pragma unroll
      for (int r = 0; r < 8; ++r)
        stg[(r + 8 * g) * 132 + t * 16 + ln] = acc[t][r];
    __syncthreads();
    float* vt = vto + (size_t)bi * KDIM * SEQ;
#pragma unroll 8
    for (int c0 = 0; c0 < KDIM; c0 += 2) {
      const float val = stg[ln * 132 + c0 + g];
      vt[(size_t)(c0 + g) * SEQ + sbase + ln] = val;
    }
  }
}

// ---------------------------------------------------------------------------
// Kernel 2: flash-style attention, one wave per (batch, 16-row q tile).
// Q tile once; K 32x128 and vT 128x32 tiles double-buffered via TDM.
// Online softmax state replicated per half-wave via __shfl_xor (wave32).
// ---------------------------------------------------------------------------
__global__ __launch_bounds__(32) void attn_kernel(
    const float* __restrict__ qf, const float* __restrict__ kf,
    const float* __restrict__ vt, const int* __restrict__ maskflag,
    float* __restrict__ out)
{
  const int lane = threadIdx.x;
  const int ln   = lane & 15;
  const int g    = lane >> 4;
  const int b    = blockIdx.y;
  const int qbase = blockIdx.x * 16;
  const int domask = maskflag[0];

  const float* qb  = qf + (size_t)b * SEQ * KDIM;
  const float* kb  = kf + (size_t)b * SEQ * KDIM;
  const float* vtb = vt + (size_t)b * KDIM * SEQ;

  __shared__ float qtile[16][132];
  __shared__ float ptile[16][36];
  __shared__ float ktile[2][32][132];
  __shared__ float vtile[2][128][36];   // vT tile: [col][s], padded

  // prologue: q tile + chunk 0 K/V in flight
  tdm_load_tile_f32(qb + (size_t)qbase * KDIM, lds_addr32(&qtile[0][0]),
                    16, 128, KDIM, 6);
  tdm_load_tile_f32(kb, lds_addr32(&ktile[0][0][0]), 32, 128, KDIM, 6);
  tdm_load_tile_f32(vtb, lds_addr32(&vtile[0][0][0]), 128, 32, SEQ, 4);

  const float MASKF  = 1.0f - (1.0f + 1.0e9f);   // -1e9 (TF mult mask)
  const float invDen = 0.022097086912079608f;    // 1/sqrt(2048)

  float mrun[8], lrun[8];
  v8f o[8] = {};
#pragma unroll
  for (int r = 0; r < 8; ++r) { mrun[r] = -1.0e30f; lrun[r] = 0.0f; }

#pragma unroll 1
  for (int ckey = 0; ckey < SEQ; ckey += 32) {
    const int cur = (ckey >> 5) & 1;
    if (ckey + 32 < SEQ) {
      tdm_load_tile_f32(kb + (size_t)(ckey + 32) * KDIM,
                        lds_addr32(&ktile[cur ^ 1][0][0]), 32, 128, KDIM, 6);
      tdm_load_tile_f32(vtb + ckey + 32,
                        lds_addr32(&vtile[cur ^ 1][0][0]), 128, 32, SEQ, 4);
      wait_tensor_le2();   // q + chunk `cur` K/V resident
    } else {
      wait_tensor_0();
    }

    // ---- scores: two 16x16 tiles, contraction over KD = 128 ----
    v8f s0 = {}, s1 = {};
#pragma unroll
    for (int kk = 0; kk < KDIM; kk += 4) {
      const v2f a  = *(const v2f*)&qtile[ln][kk + 2 * g];
      const v2f b0 = *(const v2f*)&ktile[cur][ln][kk + 2 * g];
      s0 = wmma4(a, b0, s0);
      const v2f b1 = *(const v2f*)&ktile[cur][16 + ln][kk + 2 * g];
      s1 = wmma4(a, b1, s1);
    }

    // ---- multiplicative mask + 1/sqrt(S) ----
    float e0[8], e1[8];
#pragma unroll
    for (int r = 0; r < 8; ++r) {
      const int rowi = qbase + r + 8 * g;
      const int c0 = ckey + ln;
      const int c1 = ckey + 16 + ln;
      const float f0 = (domask && (c0 >= rowi)) ? MASKF : 1.0f;
      const float f1 = (domask && (c1 >= rowi)) ? MASKF : 1.0f;
      e0[r] = s0[r] * f0 * invDen;
      e1[r] = s1[r] * f1 * invDen;
    }

    // ---- online softmax ----
    float sc[8];
#pragma unroll
    for (int r = 0; r < 8; ++r) {
      float mx = fmaxf(e0[r], e1[r]);
      mx = fmaxf(mx, __shfl_xor(mx, 1, 32));
      mx = fmaxf(mx, __shfl_xor(mx, 2, 32));
      mx = fmaxf(mx, __shfl_xor(mx, 4, 32));
      mx = fmaxf(mx, __shfl_xor(mx, 8, 32));
      const float mnew = fmaxf(mrun[r], mx);
      sc[r] = expf(mrun[r] - mnew);
      const float p0 = expf(e0[r] - mnew);
      const float p1 = expf(e1[r] - mnew);
      e0[r] = p0; e1[r] = p1;
      float rs = p0 + p1;
      rs += __shfl_xor(rs, 1, 32);
      rs += __shfl_xor(rs, 2, 32);
      rs += __shfl_xor(rs, 4, 32);
      rs += __shfl_xor(rs, 8, 32);
      lrun[r] = lrun[r] * sc[r] + rs;
      mrun[r] = mnew;
    }

    // rescale running output
#pragma unroll
    for (int t = 0; t < 8; ++t)
#pragma unroll
      for (int r = 0; r < 8; ++r) o[t][r] *= sc[r];

    // ---- probs C-fragment -> LDS -> A-fragment ----
    __syncthreads();
#pragma unroll
    for (int r = 0; r < 8; ++r) {
      ptile[r + 8 * g][ln]      = e0[r];
      ptile[r + 8 * g][16 + ln] = e1[r];
    }
    __syncthreads();

    // ---- O += P @ V (contraction over 32 chunk keys, V from vT tile) ----
#pragma unroll
    for (int kk = 0; kk < 32; kk += 4) {
      const v2f a = *(const v2f*)&ptile[ln][kk + 2 * g];
#pragma unroll
      for (int t = 0; t < 8; ++t) {
        const v2f bv = *(const v2f*)&vtile[cur][t * 16 + ln][kk + 2 * g];
        o[t] = wmma4(a, bv, o[t]);
      }
    }
  }

  // ---- normalize and store ----
  float* ob = out + (size_t)b * SEQ * KDIM;
#pragma unroll
  for (int r = 0; r < 8; ++r) {
    const float inv_l = 1.0f / lrun[r];
#pragma unroll
    for (int t = 0; t < 8; ++t) {
      ob[(size_t)(qbase + r + 8 * g) * KDIM + t * 16 + ln] = o[t][r] * inv_l;
    }
  }
}

// ---------------------------------------------------------------------------
extern "C" void kernel_launch(void* const* d_in, const int* in_sizes, int n_in,
                              void* d_out, int out_size, void* d_ws, size_t ws_size,
                              hipStream_t stream) {
  (void)in_sizes; (void)n_in; (void)out_size; (void)ws_size;

  const float* queries = (const float*)d_in[0];
  const float* keys    = (const float*)d_in[1];
  const float* values  = (const float*)d_in[2];
  const float* WQ      = (const float*)d_in[3];
  const float* WK      = (const float*)d_in[4];
  const float* WV      = (const float*)d_in[5];
  const int*   mf      = (const int*)d_in[6];
  float*       out     = (float*)d_out;

  // workspace: qf, kf (row-major), vT (transposed), 3x Wt
  float* qf  = (float*)d_ws;                               // 4 MB
  float* kf  = qf  + (size_t)BATCH * SEQ * KDIM;           // 4 MB
  float* vtT = kf  + (size_t)BATCH * SEQ * KDIM;           // 4 MB
  float* wqt = vtT + (size_t)BATCH * SEQ * KDIM;           // 512 KB
  float* wkt = wqt + (size_t)EMB * KDIM;
  float* wvt = wkt + (size_t)EMB * KDIM;

  transpose_w_kernel<<<dim3((EMB * KDIM) / 256, 3), 256, 0, stream>>>(
      WQ, WK, WV, wqt, wkt, wvt);

  proj_kernel<<<dim3((BATCH * SEQ) / 16, 3), 32, 0, stream>>>(
      queries, keys, values, wqt, wkt, wvt, qf, kf, vtT);

  attn_kernel<<<dim3(SEQ / 16, BATCH), 32, 0, stream>>>(qf, kf, vtT, mf, out);
}